// DynamicAdjSparse_84250078478505
// MI455X (gfx1250) — compile-verified
//
#include <hip/hip_runtime.h>
#include <math.h>

#define ALPHA_W 0.005f

#ifndef __has_builtin
#define __has_builtin(x) 0
#endif

#if __has_builtin(__builtin_amdgcn_global_load_async_to_lds_b128) && \
    __has_builtin(__builtin_amdgcn_global_store_async_from_lds_b128)
#define USE_ASYNC_LDS 1
#else
#define USE_ASYNC_LDS 0
#endif

// Async-LDS DMA builtins take pointers to int32x4 (b128 payload) qualified
// with the global (1) / LDS (3) address spaces (confirmed by clang diagnostics).
typedef int v4i __attribute__((ext_vector_type(4)));
typedef __attribute__((address_space(1))) v4i* gptr4;  // global b128
typedef __attribute__((address_space(3))) v4i* lptr4;  // LDS b128

__device__ __forceinline__ void wait_async0() {
  asm volatile("s_wait_asynccnt 0" ::: "memory");
}

#define TPB  256
#define FPT  16              // floats per thread
#define TILE (TPB * FPT)     // 4096 floats (16 KB LDS) per workgroup
#define BY   4               // batch fan-out split across gridDim.y

// ---------------------------------------------------------------------------
// Kernel 1: out[b,:,:] = A_base for all b.
// Stage a 16KB tile of A in LDS via async DMA (global->LDS), then fan it out
// to b_per batch slices with async LDS->global stores — no VGPR roundtrip.
// Uniform per-block range check only; full tiles run the unguarded fast path.
// ---------------------------------------------------------------------------
__global__ __launch_bounds__(TPB)
void bcast_kernel(const float* __restrict__ A, float* __restrict__ out,
                  int nn /*N*N*/, int b_per) {
  __shared__ float tile[TILE];
  const int tid = threadIdx.x;
  const size_t base = (size_t)blockIdx.x * TILE;
  const int b0 = blockIdx.y * b_per;

  if (base + TILE <= (size_t)nn) {       // uniform (scalar) branch: fast path
#if USE_ASYNC_LDS
    #pragma unroll
    for (int i = 0; i < FPT / 4; ++i) {
      const int off = (i * TPB + tid) * 4;
      __builtin_amdgcn_global_load_async_to_lds_b128(
          (gptr4)(A + base + off), (lptr4)(tile + off), 0, 0);
    }
    wait_async0();                       // tile resident (own-thread bytes only)
    for (int b = b0; b < b0 + b_per; ++b) {
      float* dst = out + (size_t)b * nn + base;
      #pragma unroll
      for (int i = 0; i < FPT / 4; ++i) {
        const int off = (i * TPB + tid) * 4;
        __builtin_amdgcn_global_store_async_from_lds_b128(
            (gptr4)(dst + off), (lptr4)(tile + off), 0, 0);
      }
    }
    wait_async0();                       // drain ASYNCcnt before LDS released
#else
    (void)tile;
    #pragma unroll
    for (int i = 0; i < FPT / 4; ++i) {
      const int off = (i * TPB + tid) * 4;
      const float4 v = *(const float4*)(A + base + off);
      for (int b = b0; b < b0 + b_per; ++b)
        *(float4*)(out + (size_t)b * nn + base + off) = v;
    }
#endif
  } else {                               // tail tile (not taken at ref shapes)
    for (int i = 0; i < FPT; ++i) {
      const size_t idx = base + (size_t)i * TPB + tid;
      if (idx < (size_t)nn) {
        const float v = A[idx];
        for (int b = b0; b < b0 + b_per; ++b)
          out[(size_t)b * nn + idx] = v;
      }
    }
  }
}

// ---------------------------------------------------------------------------
// Kernel 2: out[b, src[e], dst[e]] += 0.005 * sigmoid(clip(gate[b,e],-3,3))
// One thread per (b, e); gate loads coalesced; fire-and-forget f32 atomics
// (STOREcnt) into the L2-resident output; duplicate edges handled by the atomic.
// ---------------------------------------------------------------------------
__global__ __launch_bounds__(TPB)
void scatter_kernel(const int* __restrict__ ei, const float* __restrict__ gates,
                    float* out, int E, int nn, int N) {
  const int e = blockIdx.x * TPB + threadIdx.x;
  const int b = blockIdx.y;
  if (e >= E) return;

  float g = gates[(size_t)b * E + e];
  g = fminf(3.0f, fmaxf(-3.0f, g));
  const float w = ALPHA_W / (1.0f + expf(-g));

  const int s = ei[e];       // edge_index[0, e]
  const int d = ei[E + e];   // edge_index[1, e]
  atomicAdd(out + (size_t)b * nn + (size_t)s * N + (size_t)d, w);
}

// ---------------------------------------------------------------------------
extern "C" void kernel_launch(void* const* d_in, const int* in_sizes, int n_in,
                              void* d_out, int out_size, void* d_ws, size_t ws_size,
                              hipStream_t stream) {
  (void)n_in; (void)out_size; (void)d_ws; (void)ws_size;

  const float* A     = (const float*)d_in[0];   // [N, N]
  const int*   ei    = (const int*)d_in[1];     // [2, E]
  const float* gates = (const float*)d_in[2];   // [B, E]
  float*       out   = (float*)d_out;           // [B, N, N]

  const int nn = in_sizes[0];        // N*N = 1,048,576
  const int E  = in_sizes[1] / 2;    // 32768
  const int B  = in_sizes[2] / E;    // 32
  const int N  = (int)(sqrt((double)nn) + 0.5);  // 1024

  // Phase 1: broadcast A_base into every batch slice (fully initializes d_out).
  const int by    = (B % BY == 0) ? BY : 1;      // 4-way batch split when even
  const int b_per = B / by;
  dim3 g1((unsigned)((nn + TILE - 1) / TILE), (unsigned)by);  // 256 x 4 blocks
  bcast_kernel<<<g1, TPB, 0, stream>>>(A, out, nn, b_per);

  // Phase 2: scatter-add the gated edge weights.
  dim3 g2((unsigned)((E + TPB - 1) / TPB), (unsigned)B);      // 128 x 32 blocks
  scatter_kernel<<<g2, TPB, 0, stream>>>(ei, gates, out, E, nn, N);
}